// rpn_target_anchors_createor_26302379720849
// MI455X (gfx1250) — compile-verified
//
#include <hip/hip_runtime.h>
#include <hip/hip_bf16.h>
#include <stdint.h>

// ---------------------------------------------------------------------------
// RPN target/anchor creator for MI455X (gfx1250).
// Memory-bound workload: WMMA does not apply (IoU = min/max algebra, not a
// matmul). CDNA5 paths used instead:
//   * TDM tensor_load_to_lds (inline asm, D# built in SGPRs) stages the
//     128x4 f32 GT-box tile into LDS, waited with s_wait_tensorcnt.
//   * wave32-sized blocks, b128 vector loads/stores.
// Sampling emulated with deterministic per-index hash priorities + exact
// two-level (16b+16b) radix select via 64K-bin histograms in scratch.
// ---------------------------------------------------------------------------

#define HIST_BINS 65536
#define MAX_POS_LIMIT 128u
#define TOTAL_LIMIT 256u

// scalar slots in workspace
#define S_POS_BINT     0
#define S_POS_KPRIME   1
#define S_POS_GREATER  2
#define S_POS_THR      3
#define S_POS_NUM      4
#define S_NEG_LIMIT    5
#define S_NEG_BINT     6
#define S_NEG_KPRIME   7
#define S_NEG_GREATER  8
#define S_NEG_THR      9
#define S_NEG_NUM      10
#define S_NUM_EX       11
#define S_INV          12
#define S_POS_KEEPNONE 13
#define S_NEG_KEEPNONE 14
#define S_NSLOTS       16

typedef __attribute__((ext_vector_type(4))) unsigned int v4u;
typedef __attribute__((ext_vector_type(8))) unsigned int v8u;

__device__ __forceinline__ unsigned hash_u32(unsigned x, unsigned seed) {
    x ^= seed * 0x9E3779B9u;
    x *= 0x85EBCA6Bu; x ^= x >> 13;
    x *= 0xC2B2AE35u; x ^= x >> 16;
    x += seed;
    x ^= x >> 15; x *= 0x2C1B3C6Du; x ^= x >> 12;
    return x;
}

// ---------------------------------------------------------------------------
__global__ void rpn_zero_kernel(unsigned* __restrict__ p, int n) {
    int i = blockIdx.x * blockDim.x + threadIdx.x;
    if (i < n) p[i] = 0u;
}

// ---------------------------------------------------------------------------
// Main kernel: TDM-stage gt tile to LDS, then per-anchor IoU max/argmax,
// encode, thresholds, priority keys + hi-16 histograms.
__global__ __launch_bounds__(256)
void rpn_main_kernel(const float4* __restrict__ anchors,
                     const float*  __restrict__ gt,      // M*4 floats
                     const float*  __restrict__ ishape,  // {H, W}
                     float*        __restrict__ targets, // [N,4]
                     unsigned*     __restrict__ posKeys,
                     unsigned*     __restrict__ negKeys,
                     unsigned*     __restrict__ histPos,
                     unsigned*     __restrict__ histNeg,
                     int N, int M)
{
    __shared__ float4 sgt[128];   // only LDS object -> workgroup LDS offset 0

    // --- CDNA5 Tensor Data Mover: DMA the GT tile (M*16 bytes) into LDS. ---
    // One wave per workgroup issues it (TDM is a scalar/wave op; other waves
    // skip via execz branch). D# group0/group1 built per CDNA5 ISA.
    if (threadIdx.x < 32) {
        unsigned long long ga = (unsigned long long)(const void*)gt;
        unsigned nelem = (unsigned)(M * 4);      // f32 elements in the tile
        v4u g0;
        g0[0] = 1u;                               // count=1, user descriptor
        g0[1] = 0u;                               // lds_addr = 0 (sgt base)
        g0[2] = (unsigned)(ga & 0xFFFFFFFFull);   // global_addr[31:0]
        g0[3] = (unsigned)((ga >> 32) & 0x01FFFFFFull) | (2u << 30); // type=2
        v8u g1;
        g1[0] = 0x00020000u;                      // data_size=2 (4B), no mask
        g1[1] = (nelem & 0xFFFFu) << 16;          // tensor_dim0[15:0]
        g1[2] = ((nelem >> 16) & 0xFFFFu) | (1u << 16); // dim0 hi | tensor_dim1=1
        g1[3] = (nelem & 0xFFFFu) << 16;          // tile_dim0 = nelem
        g1[4] = 1u;                               // tile_dim1 = 1
        g1[5] = nelem;                            // tensor_dim0_stride[31:0]
        g1[6] = 0u;
        g1[7] = 0u;
        asm volatile("tensor_load_to_lds %0, %1" :: "s"(g0), "s"(g1) : "memory");
        __builtin_amdgcn_s_wait_tensorcnt(0);
    }
    __syncthreads();

    int i = blockIdx.x * blockDim.x + threadIdx.x;
    if (i >= N) return;

    float H = ishape[0];
    float W = ishape[1];
    float4 a = anchors[i];

    bool inside = (a.x >= 0.0f) && (a.y >= 0.0f) && (a.z <= W) && (a.w <= H);

    float aw = a.z - a.x;
    float ah = a.w - a.y;
    float areaA = aw * ah;

    float best = -1.0f;
    int bidx = 0;
#pragma unroll 4
    for (int m = 0; m < M; ++m) {
        float4 g = sgt[m];
        float ix = fminf(a.z, g.z) - fmaxf(a.x, g.x);
        float iy = fminf(a.w, g.w) - fmaxf(a.y, g.y);
        ix = fmaxf(ix, 0.0f);
        iy = fmaxf(iy, 0.0f);
        float inter = ix * iy;
        float areaG = (g.z - g.x) * (g.w - g.y);
        float uni   = fmaxf(areaA + areaG - inter, 1e-8f);
        float iou   = inter / uni;
        if (iou > best) { best = iou; bidx = m; }   // first-max, like argmax
    }

    // encode vs matched gt
    float4 g = sgt[bidx];
    float aw2 = fmaxf(aw, 1e-6f);
    float ah2 = fmaxf(ah, 1e-6f);
    float acx = a.x + 0.5f * aw;
    float acy = a.y + 0.5f * ah;
    float gw  = g.z - g.x;
    float gh  = g.w - g.y;
    float gcx = g.x + 0.5f * gw;
    float gcy = g.y + 0.5f * gh;
    float4 t;
    t.x = (gcx - acx) / aw2;
    t.y = (gcy - acy) / ah2;
    t.z = __logf(fmaxf(gw, 1e-6f) / aw2);
    t.w = __logf(fmaxf(gh, 1e-6f) / ah2);
    if (!inside) { t.x = 0.f; t.y = 0.f; t.z = 0.f; t.w = 0.f; }
    ((float4*)targets)[i] = t;

    bool pos = inside && (best >= 0.7f);
    bool neg = inside && (best < 0.3f);

    unsigned pk = pos ? (hash_u32((unsigned)i, 0xA511E9B3u) | 1u) : 0u;
    unsigned nk = neg ? (hash_u32((unsigned)i, 0x3C6EF372u) | 1u) : 0u;
    posKeys[i] = pk;
    negKeys[i] = nk;
    if (pk) atomicAdd(&histPos[pk >> 16], 1u);
    if (nk) atomicAdd(&histNeg[nk >> 16], 1u);
}

// ---------------------------------------------------------------------------
// Pass 1 of radix select: find the hi-16 bin where cumulative-from-top count
// crosses `limit`. Single block of 1024 threads.
__global__ void rpn_scan_hi_kernel(const unsigned* __restrict__ hist,
                                   unsigned* __restrict__ S,
                                   int use_slot_limit,
                                   int slotBinT, int slotKprime, int slotGreater,
                                   int slotThr, int slotNum, int slotKeepNone)
{
    __shared__ unsigned chunk[1024];
    int t = threadIdx.x;
    unsigned s = 0;
    int base = t * 64;
    for (int b = base; b < base + 64; ++b) s += hist[b];
    chunk[t] = s;
    __syncthreads();
    if (t == 0) {
        unsigned limit = use_slot_limit ? S[S_NEG_LIMIT] : MAX_POS_LIMIT;
        if (limit == 0u) {
            S[slotKeepNone] = 1u;
            S[slotBinT] = 0u; S[slotKprime] = 0u; S[slotGreater] = 0u;
            S[slotThr] = 0xFFFFFFFFu; S[slotNum] = 0u;
            return;
        }
        S[slotKeepNone] = 0u;
        unsigned cum = 0u;
        int T = -1;
        unsigned kprime = 0u;
        for (int c = 1023; c >= 0; --c) {
            if (cum + chunk[c] >= limit) {
                for (int b = c * 64 + 63; b >= c * 64; --b) {
                    unsigned h = hist[b];
                    if (cum + h >= limit) { T = b; kprime = limit - cum; break; }
                    cum += h;
                }
                break;
            }
            cum += chunk[c];
        }
        if (T < 0) {
            // fewer eligible than limit -> keep everything (thr=1 passes all
            // nonzero keys)
            S[slotBinT] = 0xFFFFFFFFu;
            S[slotKprime] = 0u;
            S[slotGreater] = cum;
            S[slotThr] = 1u;
            S[slotNum] = cum;
        } else {
            S[slotBinT] = (unsigned)T;
            S[slotKprime] = kprime;
            S[slotGreater] = cum;
        }
    }
}

// ---------------------------------------------------------------------------
// Build lo-16 histogram of keys falling in the threshold bin.
__global__ void rpn_lowhist_kernel(const unsigned* __restrict__ keys,
                                   unsigned* __restrict__ histLow,
                                   const unsigned* __restrict__ S,
                                   int slotBinT, int N)
{
    unsigned binT = S[slotBinT];
    if (binT == 0xFFFFFFFFu) return;   // keep-all: no refinement needed
    int i = blockIdx.x * blockDim.x + threadIdx.x;
    if (i >= N) return;
    unsigned k = keys[i];
    if (k && (k >> 16) == binT) atomicAdd(&histLow[k & 0xFFFFu], 1u);
}

// ---------------------------------------------------------------------------
// Pass 2: resolve exact 32-bit threshold inside the hi bin; also compute
// phase-dependent derived scalars (neg budget / 1/num_ex). Zeros histLow for
// reuse on the way out.
__global__ void rpn_scan_low_kernel(unsigned* __restrict__ histLow,
                                    unsigned* __restrict__ S,
                                    int slotBinT, int slotKprime, int slotGreater,
                                    int slotThr, int slotNum, int slotKeepNone,
                                    int phase)
{
    __shared__ unsigned chunk[1024];
    int t = threadIdx.x;
    unsigned s = 0;
    int base = t * 64;
    for (int b = base; b < base + 64; ++b) s += histLow[b];
    chunk[t] = s;
    __syncthreads();
    if (t == 0) {
        unsigned binT = S[slotBinT];
        if (S[slotKeepNone] == 0u && binT != 0xFFFFFFFFu) {
            unsigned kprime = S[slotKprime];
            unsigned cum = 0u;
            unsigned kept = 0u;
            int L = -1;
            for (int c = 1023; c >= 0; --c) {
                if (cum + chunk[c] >= kprime) {
                    for (int b = c * 64 + 63; b >= c * 64; --b) {
                        unsigned h = histLow[b];
                        if (cum + h >= kprime) { L = b; kept = cum + h; break; }
                        cum += h;
                    }
                    break;
                }
                cum += chunk[c];
            }
            if (L < 0) { L = 0; kept = cum; }
            S[slotThr] = (binT << 16) | (unsigned)L;
            S[slotNum] = S[slotGreater] + kept;
        }
        if (phase == 0) {
            unsigned np = S[S_POS_NUM];
            S[S_NEG_LIMIT] = (np >= TOTAL_LIMIT) ? 0u : (TOTAL_LIMIT - np);
        } else {
            unsigned ne = S[S_POS_NUM] + S[S_NEG_NUM];
            S[S_NUM_EX] = ne;
            float denom = (float)(ne > 0u ? ne : 1u);
            S[S_INV] = __float_as_uint(1.0f / denom);
        }
    }
    __syncthreads();
    for (int b = base; b < base + 64; ++b) histLow[b] = 0u;  // reuse for neg
}

// ---------------------------------------------------------------------------
__global__ __launch_bounds__(256)
void rpn_finalize_kernel(const unsigned* __restrict__ posKeys,
                         const unsigned* __restrict__ negKeys,
                         const unsigned* __restrict__ S,
                         float* __restrict__ labels,
                         float* __restrict__ w_in,
                         float* __restrict__ w_out,
                         int N)
{
    int i = blockIdx.x * blockDim.x + threadIdx.x;
    if (i >= N) return;
    unsigned pk = posKeys[i];
    unsigned nk = negKeys[i];
    unsigned pThr = S[S_POS_THR], nThr = S[S_NEG_THR];
    bool pKeep = pk && (S[S_POS_KEEPNONE] == 0u) && (pk >= pThr);
    bool nKeep = nk && (S[S_NEG_KEEPNONE] == 0u) && (nk >= nThr);
    float lab = pKeep ? 1.0f : (nKeep ? 0.0f : -1.0f);
    labels[i] = lab;
    float wi = pKeep ? 1.0f : 0.0f;
    float wo = (lab >= 0.0f) ? __uint_as_float(S[S_INV]) : 0.0f;
    float4 wiv; wiv.x = wi; wiv.y = wi; wiv.z = wi; wiv.w = wi;
    float4 wov; wov.x = wo; wov.y = wo; wov.z = wo; wov.w = wo;
    ((float4*)w_in)[i]  = wiv;
    ((float4*)w_out)[i] = wov;
}

// ---------------------------------------------------------------------------
extern "C" void kernel_launch(void* const* d_in, const int* in_sizes, int n_in,
                              void* d_out, int out_size, void* d_ws, size_t ws_size,
                              hipStream_t stream) {
    (void)n_in; (void)out_size; (void)ws_size;

    const float4* anchors = (const float4*)d_in[0];
    const float*  gt      = (const float*)d_in[1];
    const float*  ishape  = (const float*)d_in[2];

    int N = in_sizes[0] / 4;
    int M = in_sizes[1] / 4;
    if (M > 128) M = 128;

    float* out     = (float*)d_out;
    float* labels  = out;                       // [N]
    float* targets = out + (size_t)N;           // [N,4]
    float* w_in    = out + (size_t)5 * N;       // [N,4]
    float* w_out   = out + (size_t)9 * N;       // [N,4]

    unsigned* ws      = (unsigned*)d_ws;
    unsigned* posKeys = ws;                               // N
    unsigned* negKeys = ws + (size_t)N;                   // N
    unsigned* histPos = ws + (size_t)2 * N;               // 64K
    unsigned* histNeg = histPos + HIST_BINS;              // 64K
    unsigned* histLow = histNeg + HIST_BINS;              // 64K (shared)
    unsigned* S       = histLow + HIST_BINS;              // scalars

    int blocks = (N + 255) / 256;
    int zn = 3 * HIST_BINS + S_NSLOTS;

    rpn_zero_kernel<<<(zn + 255) / 256, 256, 0, stream>>>(histPos, zn);

    rpn_main_kernel<<<blocks, 256, 0, stream>>>(anchors, gt, ishape, targets,
                                                posKeys, negKeys, histPos,
                                                histNeg, N, M);

    // positive selection: top-128 by key
    rpn_scan_hi_kernel<<<1, 1024, 0, stream>>>(histPos, S, 0,
        S_POS_BINT, S_POS_KPRIME, S_POS_GREATER, S_POS_THR, S_POS_NUM, S_POS_KEEPNONE);
    rpn_lowhist_kernel<<<blocks, 256, 0, stream>>>(posKeys, histLow, S, S_POS_BINT, N);
    rpn_scan_low_kernel<<<1, 1024, 0, stream>>>(histLow, S,
        S_POS_BINT, S_POS_KPRIME, S_POS_GREATER, S_POS_THR, S_POS_NUM, S_POS_KEEPNONE, 0);

    // negative selection: top-(256 - num_pos) by key
    rpn_scan_hi_kernel<<<1, 1024, 0, stream>>>(histNeg, S, 1,
        S_NEG_BINT, S_NEG_KPRIME, S_NEG_GREATER, S_NEG_THR, S_NEG_NUM, S_NEG_KEEPNONE);
    rpn_lowhist_kernel<<<blocks, 256, 0, stream>>>(negKeys, histLow, S, S_NEG_BINT, N);
    rpn_scan_low_kernel<<<1, 1024, 0, stream>>>(histLow, S,
        S_NEG_BINT, S_NEG_KPRIME, S_NEG_GREATER, S_NEG_THR, S_NEG_NUM, S_NEG_KEEPNONE, 1);

    rpn_finalize_kernel<<<blocks, 256, 0, stream>>>(posKeys, negKeys, S,
                                                    labels, w_in, w_out, N);
}